// Loss_13683765805470
// MI455X (gfx1250) — compile-verified
//
#include <hip/hip_runtime.h>

// Problem sizes (match reference)
#define B_     4096
#define S_     512
#define T_     3
#define E_     32
#define CENT_  4
#define CINT_  10

// CRF parallel-scan configuration
#define NCHUNK 16
#define CHUNK  32                 // S_ / NCHUNK
#define STG    4                  // steps staged per async refill (48B em + 16B mask)
#define NREFILL (CHUNK / STG)     // 8
#define BLK    256
#define NWAVES (BLK / 32)         // 8

#define ENT_BLOCKS ((B_ * E_) / BLK)   // 512
#define INT_BLOCKS (B_ / BLK)          // 16

#define NEGINF -1e30f

typedef __attribute__((ext_vector_type(2))) float v2f;
typedef __attribute__((ext_vector_type(8))) float v8f;

__device__ __forceinline__ float lse3f(float a, float b, float c) {
  float m = fmaxf(a, fmaxf(b, c));
  return m + __logf(__expf(a - m) + __expf(b - m) + __expf(c - m));
}

// ---------------------------------------------------------------------------
// Wave-wide exact f32 sum via one V_WMMA_F32_16X16X4_F32 with B = ones(4x16):
// D[m][j] = sum_k A[m][k]  (replicated over j). Each lane contributes 2 floats
// (A slots K0/K1 on lanes 0-15, K2/K3 on lanes 16-31 — placement irrelevant
// for a total sum). Wave total = (per-lane sum of C rows)@lane0 + @lane16.
// Requires EXEC = all ones (callers are fully-populated, non-divergent blocks).
// ---------------------------------------------------------------------------
__device__ __forceinline__ float wmma_wave_sum(float x0, float x1) {
  v2f a;    a[0] = x0;  a[1] = x1;
  v2f ones; ones[0] = 1.f; ones[1] = 1.f;
  v8f c = {};
  c = __builtin_amdgcn_wmma_f32_16x16x4_f32(false, a, false, ones, (short)0, c,
                                            false, false);
  float h = ((c[0] + c[1]) + (c[2] + c[3])) + ((c[4] + c[5]) + (c[6] + c[7]));
  return __shfl(h, 0, 32) + __shfl(h, 16, 32);
}

// ---------------------------------------------------------------------------
// Kernel 1: per-(row,chunk) 3x3 log-semiring matrix via serial scan of CHUNK
// steps, with double-buffered async global->LDS staging of emissions+mask.
// ---------------------------------------------------------------------------
__global__ __launch_bounds__(BLK)
void crf_chunk_kernel(const float* __restrict__ em,
                      const int*   __restrict__ mask,
                      const float* __restrict__ trans,
                      float*       __restrict__ mats)
{
  __shared__ unsigned char stage[BLK * 128];   // 128B per thread, 2 buffers of 64B

  const int gid = blockIdx.x * BLK + threadIdx.x;
  const int b   = gid >> 4;              // / NCHUNK
  const int c   = gid & (NCHUNK - 1);
  const int s0  = c * CHUNK;

  float tr[9];
#pragma unroll
  for (int k = 0; k < 9; ++k) tr[k] = trans[k];   // uniform -> SGPRs

  unsigned char* my = stage + threadIdx.x * 128;
#pragma unroll
  for (int k = 0; k < 8; ++k) ((float4*)my)[k] = make_float4(0.f, 0.f, 0.f, 0.f);
  // make sure the init stores land before the async DMA writes the same bytes
  asm volatile("s_wait_dscnt 0x0" ::: "memory");

  const unsigned           ldsBase = (unsigned)(unsigned long long)my;
  const unsigned long long emBase  = (unsigned long long)(em + ((size_t)b * S_ + s0) * T_);
  const unsigned long long mkBase  = (unsigned long long)(mask + ((size_t)b * S_ + s0));

  // A = semiring identity (diag 0, off-diag -inf)
  float A[9] = {0.f, NEGINF, NEGINF, NEGINF, 0.f, NEGINF, NEGINF, NEGINF, 0.f};

  auto issue = [&](int buf, int refill) {
    unsigned           l  = ldsBase + (unsigned)(buf * 64);
    unsigned long long ge = emBase + (unsigned long long)(refill * (STG * T_ * 4)); // 48B
    unsigned long long gm = mkBase + (unsigned long long)(refill * (STG * 4));      // 16B
    asm volatile("global_load_async_to_lds_b128 %0, %1, off" :: "v"(l       ), "v"(ge        ) : "memory");
    asm volatile("global_load_async_to_lds_b128 %0, %1, off" :: "v"(l + 16u ), "v"(ge + 16ull) : "memory");
    asm volatile("global_load_async_to_lds_b128 %0, %1, off" :: "v"(l + 32u ), "v"(ge + 32ull) : "memory");
    asm volatile("global_load_async_to_lds_b128 %0, %1, off" :: "v"(l + 48u ), "v"(gm        ) : "memory");
  };

  issue(0, 0);
  for (int r = 0; r < NREFILL; ++r) {
    if (r + 1 < NREFILL) {
      issue((r + 1) & 1, r + 1);                        // prefetch next buffer
      asm volatile("s_wait_asynccnt 0x4" ::: "memory"); // current buffer's 4 ops done
    } else {
      asm volatile("s_wait_asynccnt 0x0" ::: "memory");
    }
    const unsigned char* bp = my + (r & 1) * 64;
    float4 e0 = *(const float4*)(bp + 0);
    float4 e1 = *(const float4*)(bp + 16);
    float4 e2 = *(const float4*)(bp + 32);
    int4   mi = *(const int4*)  (bp + 48);
    float emv[12] = {e0.x, e0.y, e0.z, e0.w, e1.x, e1.y, e1.z, e1.w, e2.x, e2.y, e2.z, e2.w};
    int   mk[4]   = {mi.x, mi.y, mi.z, mi.w};
#pragma unroll
    for (int k = 0; k < STG; ++k) {
      const int s = s0 + r * STG + k;
      if (mk[k] && (s >= 1)) {                 // step 0 is alpha0, handled in combine
        float ej0 = emv[k * 3 + 0], ej1 = emv[k * 3 + 1], ej2 = emv[k * 3 + 2];
        float N[9];
#pragma unroll
        for (int i = 0; i < 3; ++i) {
          float a0 = A[i * 3 + 0], a1 = A[i * 3 + 1], a2 = A[i * 3 + 2];
          N[i * 3 + 0] = lse3f(a0 + tr[0], a1 + tr[3], a2 + tr[6]) + ej0;
          N[i * 3 + 1] = lse3f(a0 + tr[1], a1 + tr[4], a2 + tr[7]) + ej1;
          N[i * 3 + 2] = lse3f(a0 + tr[2], a1 + tr[5], a2 + tr[8]) + ej2;
        }
#pragma unroll
        for (int q = 0; q < 9; ++q) A[q] = N[q];
      }
    }
  }

  float* o = mats + ((size_t)b * NCHUNK + c) * 9;
#pragma unroll
  for (int q = 0; q < 9; ++q) o[q] = A[q];
}

// ---------------------------------------------------------------------------
// Kernel 2: per-row combine (log-partition from chunk matrices) + gold score.
// ---------------------------------------------------------------------------
__global__ __launch_bounds__(BLK)
void crf_combine_kernel(const float* __restrict__ em,
                        const int*   __restrict__ mask,
                        const int*   __restrict__ tags,
                        const float* __restrict__ start_t,
                        const float* __restrict__ end_t,
                        const float* __restrict__ trans,
                        const float* __restrict__ mats,
                        float*       __restrict__ rowv)
{
  const int b = blockIdx.x * BLK + threadIdx.x;
  const float* emb = em   + (size_t)b * S_ * T_;
  const int*   mkb = mask + (size_t)b * S_;
  const int*   tgb = tags + (size_t)b * S_;

  // log-partition: v = (start + em0) folded through the 16 chunk matrices
  float v0 = start_t[0] + emb[0];
  float v1 = start_t[1] + emb[1];
  float v2 = start_t[2] + emb[2];
  const float* M = mats + (size_t)b * NCHUNK * 9;
#pragma unroll
  for (int cc = 0; cc < NCHUNK; ++cc) {
    const float* P = M + cc * 9;
    float n0 = lse3f(v0 + P[0], v1 + P[3], v2 + P[6]);
    float n1 = lse3f(v0 + P[1], v1 + P[4], v2 + P[7]);
    float n2 = lse3f(v0 + P[2], v1 + P[5], v2 + P[8]);
    v0 = n0; v1 = n1; v2 = n2;
  }
  const float denom = lse3f(v0 + end_t[0], v1 + end_t[1], v2 + end_t[2]);

  // gold-path score (mask is a prefix; 'last' tracks the final valid tag)
  int   tprev = tgb[0];
  float score = start_t[tprev] + emb[tprev];
  int   last  = tprev;
  for (int sb = 1; sb < S_; sb += 64) {
    const int pf = (sb + 64 < S_) ? sb + 64 : S_ - 1;
    __builtin_prefetch(tgb + pf, 0, 1);      // -> global_prefetch_b8
    __builtin_prefetch(mkb + pf, 0, 1);
    const int se = (sb + 64 < S_) ? sb + 64 : S_;
    for (int s = sb; s < se; ++s) {
      const int m = mkb[s];
      const int t = tgb[s];
      if (m) {
        score += trans[tprev * 3 + t] + emb[s * 3 + t];
        last = t;
      }
      tprev = t;
    }
  }
  score += end_t[last];
  rowv[b] = denom - score;
}

// ---------------------------------------------------------------------------
// Kernel 3: entity cross-entropy (4 classes, ignore_index = 0).
// Block partials via WMMA wave reductions.
// ---------------------------------------------------------------------------
__global__ __launch_bounds__(BLK)
void entity_ce_kernel(const float* __restrict__ logit,
                      const int*   __restrict__ lab,
                      float*       __restrict__ part)
{
  const int r = blockIdx.x * BLK + threadIdx.x;
  const float4 x = ((const float4*)logit)[r];
  const int L = lab[r];
  float m   = fmaxf(fmaxf(x.x, x.y), fmaxf(x.z, x.w));
  float lse = m + __logf(__expf(x.x - m) + __expf(x.y - m) + __expf(x.z - m) + __expf(x.w - m));
  float xl    = (L == 0) ? x.x : (L == 1) ? x.y : (L == 2) ? x.z : x.w;
  float valid = (L != 0) ? 1.f : 0.f;
  float nll   = (lse - xl) * valid;

  const float wn = wmma_wave_sum(nll,   0.f);
  const float wv = wmma_wave_sum(valid, 0.f);

  __shared__ float pA[NWAVES], pB[NWAVES];
  const int wid = threadIdx.x >> 5, lane = threadIdx.x & 31;
  if (lane == 0) { pA[wid] = wn; pB[wid] = wv; }
  __syncthreads();
  if (threadIdx.x == 0) {
    float s1 = 0.f, s2 = 0.f;
#pragma unroll
    for (int w = 0; w < NWAVES; ++w) { s1 += pA[w]; s2 += pB[w]; }
    part[2 * blockIdx.x] = s1; part[2 * blockIdx.x + 1] = s2;
  }
}

// ---------------------------------------------------------------------------
// Kernel 4: intent cross-entropy (10 classes), WMMA wave-reduced partials.
// ---------------------------------------------------------------------------
__global__ __launch_bounds__(BLK)
void intent_ce_kernel(const float* __restrict__ logit,
                      const int*   __restrict__ lab,
                      float*       __restrict__ part)
{
  const int r = blockIdx.x * BLK + threadIdx.x;
  const float* x = logit + (size_t)r * CINT_;
  float xv[CINT_];
#pragma unroll
  for (int k = 0; k < CINT_; ++k) xv[k] = x[k];
  float m = xv[0];
#pragma unroll
  for (int k = 1; k < CINT_; ++k) m = fmaxf(m, xv[k]);
  float s = 0.f;
#pragma unroll
  for (int k = 0; k < CINT_; ++k) s += __expf(xv[k] - m);
  const int L = lab[r];
  const float nll = (m + __logf(s)) - x[L];

  const float wsum = wmma_wave_sum(nll, 0.f);

  __shared__ float pA[NWAVES];
  const int wid = threadIdx.x >> 5, lane = threadIdx.x & 31;
  if (lane == 0) pA[wid] = wsum;
  __syncthreads();
  if (threadIdx.x == 0) {
    float s1 = 0.f;
#pragma unroll
    for (int w = 0; w < NWAVES; ++w) s1 += pA[w];
    part[blockIdx.x] = s1;
  }
}

// ---------------------------------------------------------------------------
// Kernel 5: deterministic single-block finalize (WMMA wave reductions).
// ---------------------------------------------------------------------------
__global__ __launch_bounds__(BLK)
void finalize_kernel(const float* __restrict__ rowv,
                     const float* __restrict__ entp,
                     const float* __restrict__ intp,
                     float*       __restrict__ out)
{
  __shared__ float pA[NWAVES], pB[NWAVES];
  const int t = threadIdx.x, wid = t >> 5, lane = t & 31;

  float sum1 = 0.f, sum2 = 0.f, cnt2 = 0.f, sum3 = 0.f;

  // phase 1: CRF row losses (B_ values)
  float a = 0.f;
  for (int i = t; i < B_; i += BLK) a += rowv[i];
  float wa = wmma_wave_sum(a, 0.f);
  if (lane == 0) pA[wid] = wa;
  __syncthreads();
  if (t == 0) {
#pragma unroll
    for (int w = 0; w < NWAVES; ++w) sum1 += pA[w];
  }
  __syncthreads();

  // phase 2: entity partial sums + valid counts
  float se = 0.f, ce = 0.f;
  for (int i = t; i < ENT_BLOCKS; i += BLK) { se += entp[2 * i]; ce += entp[2 * i + 1]; }
  float wse = wmma_wave_sum(se, 0.f);
  float wce = wmma_wave_sum(ce, 0.f);
  if (lane == 0) { pA[wid] = wse; pB[wid] = wce; }
  __syncthreads();
  if (t == 0) {
#pragma unroll
    for (int w = 0; w < NWAVES; ++w) { sum2 += pA[w]; cnt2 += pB[w]; }
  }
  __syncthreads();

  // phase 3: intent partial sums
  float si = 0.f;
  for (int i = t; i < INT_BLOCKS; i += BLK) si += intp[i];
  float wsi = wmma_wave_sum(si, 0.f);
  if (lane == 0) pA[wid] = wsi;
  __syncthreads();

  if (t == 0) {
#pragma unroll
    for (int w = 0; w < NWAVES; ++w) sum3 += pA[w];
    float l1 = sum1 / (float)B_;
    float l2 = sum2 / fmaxf(cnt2, 1.f);
    float l3 = sum3 / (float)B_;
    out[0] = (l1 + l2 + l3) / 3.f;
    out[1] = l1; out[2] = l2; out[3] = l3;
  }
}

// ---------------------------------------------------------------------------
extern "C" void kernel_launch(void* const* d_in, const int* in_sizes, int n_in,
                              void* d_out, int out_size, void* d_ws, size_t ws_size,
                              hipStream_t stream) {
  (void)in_sizes; (void)n_in; (void)out_size; (void)ws_size;
  const float* em        = (const float*)d_in[0];
  const int*   mask      = (const int*)  d_in[1];
  const int*   tags      = (const int*)  d_in[2];
  const float* ent_logit = (const float*)d_in[3];
  const int*   ent_lab   = (const int*)  d_in[4];
  const float* int_logit = (const float*)d_in[5];
  const int*   int_lab   = (const int*)  d_in[6];
  const float* start_t   = (const float*)d_in[7];
  const float* end_t     = (const float*)d_in[8];
  const float* trans     = (const float*)d_in[9];
  float* out = (float*)d_out;

  float* ws   = (float*)d_ws;
  float* mats = ws;                                   // B_*NCHUNK*9 floats
  float* rowv = mats + (size_t)B_ * NCHUNK * 9;       // B_ floats
  float* entp = rowv + B_;                            // 2*ENT_BLOCKS floats
  float* intp = entp + 2 * ENT_BLOCKS;                // INT_BLOCKS floats

  crf_chunk_kernel  <<<(B_ * NCHUNK) / BLK, BLK, 0, stream>>>(em, mask, trans, mats);
  crf_combine_kernel<<<B_ / BLK,            BLK, 0, stream>>>(em, mask, tags, start_t, end_t,
                                                              trans, mats, rowv);
  entity_ce_kernel  <<<ENT_BLOCKS,          BLK, 0, stream>>>(ent_logit, ent_lab, entp);
  intent_ce_kernel  <<<INT_BLOCKS,          BLK, 0, stream>>>(int_logit, int_lab, intp);
  finalize_kernel   <<<1,                   BLK, 0, stream>>>(rowv, entp, intp, out);
}